// MomentNet_66554813219075
// MI455X (gfx1250) — compile-verified
//
#include <hip/hip_runtime.h>

typedef __attribute__((ext_vector_type(2))) float v2f;
typedef __attribute__((ext_vector_type(8))) float v8f;

#define BN_EPS 1e-5f

// ---------------------------------------------------------------------------
// Kernel 1: Chebyshev moment features.
// grid.x = B (32), block = 256 threads (8 wave32s on one WGP).
// feats[b, m] = (1/N) * sum_n  T_{e0}(x0) * T_{e1}(x1) * T_{e2}(x2)
// cos(k*acos(x)) == T_k(x), so the whole transcendental block collapses to
// the Chebyshev recurrence T_k = 2x*T_{k-1} - T_{k-2}.
// ---------------------------------------------------------------------------
__global__ __launch_bounds__(256) void moments_kernel(
    const float* __restrict__ x,     // (B, 3, N)
    const int*   __restrict__ exps,  // (M, 3) int32
    float*       __restrict__ feats, // (B, Kpad) row-major, tail zero-padded
    int N, int M, int Kpad)
{
    const int b   = blockIdx.x;
    const int tid = threadIdx.x;

    // [point][j*16 + k], stride 49 to spread the 64 LDS banks
    __shared__ float Ttab[256][49];

    // Monomial ownership: m = tid + i*256, up to 4 accumulators in registers.
    int   e0[4], e1[4], e2[4];
    float acc[4];
    int nown = 0;
    #pragma unroll
    for (int i = 0; i < 4; ++i) {
        int m = tid + i * 256;
        if (m < M) {
            e0[i] = exps[m * 3 + 0];
            e1[i] = exps[m * 3 + 1];
            e2[i] = exps[m * 3 + 2];
            acc[i] = 0.0f;
            nown = i + 1;
        }
    }

    const int tiles = N >> 8;  // N / 256
    for (int t = 0; t < tiles; ++t) {
        const int n = (t << 8) + tid;
        // Phase 1: build T tables for this tile's 256 points
        #pragma unroll
        for (int j = 0; j < 3; ++j) {
            float xv  = x[(b * 3 + j) * N + n];   // coalesced
            float tm2 = 1.0f, tm1 = xv;
            Ttab[tid][j * 16 + 0] = tm2;
            Ttab[tid][j * 16 + 1] = tm1;
            #pragma unroll
            for (int k = 2; k < 16; ++k) {
                float tk = 2.0f * xv * tm1 - tm2;
                Ttab[tid][j * 16 + k] = tk;
                tm2 = tm1; tm1 = tk;
            }
        }
        __syncthreads();

        // Phase 2: accumulate owned monomials over all 256 points of the tile
        for (int p = 0; p < 256; ++p) {
            const float* Tp = Ttab[p];
            #pragma unroll
            for (int i = 0; i < 4; ++i) {
                if (i < nown) {
                    acc[i] += Tp[e0[i]] * Tp[16 + e1[i]] * Tp[32 + e2[i]];
                }
            }
        }
        __syncthreads();
    }

    const float invN = 1.0f / (float)N;
    for (int i = 0; i < nown; ++i) {
        int m = tid + i * 256;
        feats[b * Kpad + m] = acc[i] * invN;
    }
    if (tid == 0) {
        for (int m = M; m < Kpad; ++m) feats[b * Kpad + m] = 0.0f;  // K padding
    }
}

// ---------------------------------------------------------------------------
// Kernel 2: transpose W (K x H, row-major) into Wt (Hpad x Kpad, row-major),
// zero-filled outside [H) x [K).  Makes the per-lane B-fragment a single
// aligned b64 load and removes all bounds checks from the GEMM inner loop.
// ---------------------------------------------------------------------------
__global__ void transpose_w_kernel(const float* __restrict__ W,
                                   float* __restrict__ Wt,
                                   int K, int H, int Kpad, int Hpad)
{
    int idx = blockIdx.x * blockDim.x + threadIdx.x;
    int total = Hpad * Kpad;
    if (idx < total) {
        int c = idx / Kpad;   // output row   = W column
        int k = idx % Kpad;   // output col   = W row
        Wt[idx] = (c < H && k < K) ? W[k * H + c] : 0.0f;
    }
}

// ---------------------------------------------------------------------------
// Kernel 3: C(32 x H) = A(32 x K) @ W(K x H) [+ bias, BN(eval), ReLU]
// One wave32 per 16-column tile; two 16-row WMMA accumulators cover M=32.
// V_WMMA_F32_16X16X4_F32 (full fp32 matrix pipe).
//
// Fragment layouts per CDNA5 ISA 7.12.2 (wave32):
//   A 16x4 : lane l -> row (l&15),   k = 2*(l>>4) + vgpr
//   B 4x16 : lane l -> col (l&15),   k = 2*(l>>4) + vgpr
//   C 16x16: lane l -> col (l&15),   row = vgpr + 8*(l>>4)
//
// Software-pipelined: next k-step's 3x global_load_b64 are issued before the
// WMMAs consume the current fragments, so the per-step s_wait only covers
// loads that already had a full iteration of latency hiding.
// ---------------------------------------------------------------------------
__global__ __launch_bounds__(32) void gemm32_bn_relu_kernel(
    const float* __restrict__ A,     // (32, lda) row-major, K zero-padded
    const float* __restrict__ Wt,    // (Hpad, Kpad) row-major (transposed W)
    const float* __restrict__ bias,  // (H)
    const float* __restrict__ g,     // (H) or nullptr
    const float* __restrict__ be,
    const float* __restrict__ rm,
    const float* __restrict__ rv,
    float*       __restrict__ out,   // (32, H) row-major
    int K, int lda, int H, int bn_relu)
{
    const int l  = threadIdx.x;       // 0..31
    const int r  = l & 15;
    const int kk = (l >> 4) * 2;
    const int c0 = blockIdx.x * 16;
    const int c  = c0 + r;            // this lane's column

    const float* pa0 = A  + r * lda        + kk;
    const float* pa1 = A  + (r + 16) * lda + kk;
    const float* pb  = Wt + c * lda        + kk;   // Kpad == lda for all layers

    v8f acc0 = {};   // rows 0..15
    v8f acc1 = {};   // rows 16..31

    // Prologue: fragments for k=0
    v2f a0 = *(const v2f*)(pa0);
    v2f a1 = *(const v2f*)(pa1);
    v2f bf = *(const v2f*)(pb);

    for (int k0 = 0; k0 < K; k0 += 4) {
        // Uniform scalar clamp keeps the tail prefetch in-bounds (its value is
        // never consumed) without any EXEC divergence.
        const int kn = (k0 + 4 < K) ? (k0 + 4) : 0;
        v2f na0 = *(const v2f*)(pa0 + kn);
        v2f na1 = *(const v2f*)(pa1 + kn);
        v2f nbf = *(const v2f*)(pb  + kn);

        acc0 = __builtin_amdgcn_wmma_f32_16x16x4_f32(
            false, a0, false, bf, (short)0, acc0, false, false);
        acc1 = __builtin_amdgcn_wmma_f32_16x16x4_f32(
            false, a1, false, bf, (short)0, acc1, false, false);

        a0 = na0; a1 = na1; bf = nbf;
    }

    // Epilogue: bias + (optional) folded BatchNorm + ReLU
    float bsv = 0.0f, scale = 1.0f, shift = 0.0f;
    if (c < H) {
        bsv = bias[c];
        if (bn_relu) {
            float s = g[c] * rsqrtf(rv[c] + BN_EPS);
            scale = s;
            shift = be[c] - rm[c] * s;
        }
    }

    const int rowbase = (l >> 4) * 8;
    #pragma unroll
    for (int v = 0; v < 8; ++v) {
        float v0 = acc0[v] + bsv;
        float v1 = acc1[v] + bsv;
        if (bn_relu) {
            v0 = fmaxf(v0 * scale + shift, 0.0f);
            v1 = fmaxf(v1 * scale + shift, 0.0f);
        }
        if (c < H) {
            int row = rowbase + v;
            out[row * H + c]        = v0;
            out[(row + 16) * H + c] = v1;
        }
    }
}

// ---------------------------------------------------------------------------
// Host-side launch
// ---------------------------------------------------------------------------
extern "C" void kernel_launch(void* const* d_in, const int* in_sizes, int n_in,
                              void* d_out, int out_size, void* d_ws, size_t ws_size,
                              hipStream_t stream)
{
    (void)n_in; (void)out_size; (void)ws_size;

    const float* x    = (const float*)d_in[0];   // (32, 3, 2048)
    const int*   exps = (const int*)  d_in[1];   // (815, 3)
    const float* W1 = (const float*)d_in[2];
    const float* b1 = (const float*)d_in[3];
    const float* g1 = (const float*)d_in[4];
    const float* be1= (const float*)d_in[5];
    const float* rm1= (const float*)d_in[6];
    const float* rv1= (const float*)d_in[7];
    const float* W2 = (const float*)d_in[8];
    const float* b2 = (const float*)d_in[9];
    const float* g2 = (const float*)d_in[10];
    const float* be2= (const float*)d_in[11];
    const float* rm2= (const float*)d_in[12];
    const float* rv2= (const float*)d_in[13];
    const float* W3 = (const float*)d_in[14];
    const float* b3 = (const float*)d_in[15];
    const float* g3 = (const float*)d_in[16];
    const float* be3= (const float*)d_in[17];
    const float* rm3= (const float*)d_in[18];
    const float* rv3= (const float*)d_in[19];
    const float* W4 = (const float*)d_in[20];
    const float* b4 = (const float*)d_in[21];
    float* out = (float*)d_out;                  // (32, 40)

    const int B = 32, N = 2048;
    const int M = in_sizes[1] / 3;               // 815
    const int K1 = (M + 3) & ~3;                 // 816 (padded feature count)
    const int H1 = 512, H2 = 256, H3 = 128, H4 = 40;
    const int H4p = 48;                          // 40 -> 3 tiles of 16

    // Workspace layout (floats)
    float* ws    = (float*)d_ws;
    float* feats = ws;                           // 32  * 816
    float* W1t   = feats + B * K1;               // 512 * 816
    float* W2t   = W1t + H1 * K1;                // 256 * 512
    float* W3t   = W2t + H2 * H1;                // 128 * 256
    float* W4t   = W3t + H3 * H2;                // 48  * 128
    float* h1    = W4t + H4p * H3;               // 32  * 512
    float* h2    = h1 + B * H1;                  // 32  * 256
    float* h3    = h2 + B * H2;                  // 32  * 128

    // 1) Chebyshev moment features (recurrence, no transcendentals)
    moments_kernel<<<B, 256, 0, stream>>>(x, exps, feats, N, M, K1);

    // 2) Transpose + zero-pad all weights
    {
        int t1 = H1 * K1;
        transpose_w_kernel<<<(t1 + 255) / 256, 256, 0, stream>>>(W1, W1t, M,  H1, K1, H1);
        int t2 = H2 * H1;
        transpose_w_kernel<<<(t2 + 255) / 256, 256, 0, stream>>>(W2, W2t, H1, H2, H1, H2);
        int t3 = H3 * H2;
        transpose_w_kernel<<<(t3 + 255) / 256, 256, 0, stream>>>(W3, W3t, H2, H3, H2, H3);
        int t4 = H4p * H3;
        transpose_w_kernel<<<(t4 + 255) / 256, 256, 0, stream>>>(W4, W4t, H3, H4, H3, H4p);
    }

    // 3) MLP: f32 WMMA GEMMs with fused bias/BN/ReLU
    gemm32_bn_relu_kernel<<<H1 / 16, 32, 0, stream>>>(
        feats, W1t, b1, g1, be1, rm1, rv1, h1, K1, K1, H1, 1);
    gemm32_bn_relu_kernel<<<H2 / 16, 32, 0, stream>>>(
        h1, W2t, b2, g2, be2, rm2, rv2, h2, H1, H1, H2, 1);
    gemm32_bn_relu_kernel<<<H3 / 16, 32, 0, stream>>>(
        h2, W3t, b3, g3, be3, rm3, rv3, h3, H2, H2, H3, 1);
    gemm32_bn_relu_kernel<<<H4p / 16, 32, 0, stream>>>(
        h3, W4t, b4, nullptr, nullptr, nullptr, nullptr, out, H3, H3, H4, 0);
}